// EmbedEdgeModel_52252572123261
// MI455X (gfx1250) — compile-verified
//
#include <hip/hip_runtime.h>
#include <hip/hip_bf16.h>
#include <stdint.h>

// CDNA5 (gfx1250) wave32 WMMA f32 16x16x4 kernel for:
//   out = mask ? relu(relu(X@W1+b1)@W2+b2) : 0,  X:[E,32], W1:[32,64], W2:[64,32]
//
// Roofline: 13.1 GFLOP vs ~411 MB HBM -> ~17.6us memory floor @ 23.3 TB/s.
// AI = 32 FLOP/B, so exact-f32 V_WMMA_F32_16X16X4_F32 keeps us at the roofline
// without the precision loss of a bf16 downcast. 64 WMMAs per 16-edge tile.

typedef __attribute__((ext_vector_type(2))) float v2f;
typedef __attribute__((ext_vector_type(8))) float v8f;

#define W1T_STRIDE 36   // 32 K + 4 pad  -> lane bank stride 36 (conflict-free b64)
#define W2T_STRIDE 68   // 64 K + 4 pad  -> lane bank stride 4  (conflict-free b64)
#define H1_STRIDE  68   // 64 K + 4 pad
#define WAVES_PER_BLOCK 8

__global__ __launch_bounds__(256)
void edge_mlp_wmma_f32(const float* __restrict__ x,          // [E,32]
                       const float* __restrict__ W1,         // [32,64] row-major
                       const float* __restrict__ b1,         // [64]
                       const float* __restrict__ W2,         // [64,32] row-major
                       const float* __restrict__ b2,         // [32]
                       const unsigned char* __restrict__ mask, // [E] (jax bool = 1B)
                       float* __restrict__ out,              // [E,32]
                       int E)
{
  // LDS: transposed weights (shared by block) + per-wave h1 transpose staging.
  __shared__ float sW1T[64 * W1T_STRIDE];                    // W1T[n][k] = W1[k][n]
  __shared__ float sW2T[32 * W2T_STRIDE];                    // W2T[n][k] = W2[k][n]
  __shared__ float sH1 [WAVES_PER_BLOCK * 16 * H1_STRIDE];   // h1[m][k], per wave

  const int tid = threadIdx.x;

  // Cooperative one-time weight transpose into LDS (weights are L2-resident).
  for (int idx = tid; idx < 32 * 64; idx += 256) {
    const int k1 = idx >> 6, n1 = idx & 63;   // W1[k1][n1]
    sW1T[n1 * W1T_STRIDE + k1] = W1[idx];
    const int k2 = idx >> 5, n2 = idx & 31;   // W2[k2][n2]
    sW2T[n2 * W2T_STRIDE + k2] = W2[idx];
  }
  __syncthreads();

  const int lane = tid & 31;
  const int wave = tid >> 5;
  const int col  = lane & 15;     // N index inside a 16-wide tile (C layout lane)
  const int hi   = lane >> 4;     // half-wave selector (K/M split per ISA layouts)
  float* myH1 = &sH1[wave * 16 * H1_STRIDE];

  // Bias pre-broadcast: C/D layout has lane == N, so per-lane splat == bias bcast.
  float b1v[4], b2v[2];
#pragma unroll
  for (int t = 0; t < 4; ++t) b1v[t] = b1[t * 16 + col];
#pragma unroll
  for (int t = 0; t < 2; ++t) b2v[t] = b2[t * 16 + col];

  const int nTiles     = (E + 15) >> 4;
  const int waveGlobal = blockIdx.x * WAVES_PER_BLOCK + wave;
  const int waveStride = gridDim.x * WAVES_PER_BLOCK;

  for (int tile = waveGlobal; tile < nTiles; tile += waveStride) {
    const int  base = tile << 4;
    const bool full = (base + 16) <= E;           // wave-uniform

    // Prefetch next tile's rows (emits global_prefetch_b8).
    {
      const int nt = tile + waveStride;
      if (nt < nTiles) {
        const float* p = x + (size_t)(nt * 16 + col) * 32 + hi * 16;
        __builtin_prefetch(p, 0, 1);
      }
    }

    // ---------------- Layer 1: [16x32] @ [32x64] ----------------
    // A layout (32-bit 16x4): lane m = col; v0/v1 hold K = 4kt + {0,1} (lo half)
    // or {2,3} (hi half)  ->  per-lane float2 global load.
    const int rowL = full ? (base + col)
                          : ((base + col) < E ? (base + col) : (E - 1));
    const float* xrow = x + (size_t)rowL * 32 + hi * 2;

    v8f acc1[4];
#pragma unroll
    for (int t = 0; t < 4; ++t) {
#pragma unroll
      for (int i = 0; i < 8; ++i) acc1[t][i] = b1v[t];
    }

#pragma unroll
    for (int kt = 0; kt < 8; ++kt) {
      const v2f a = *(const v2f*)(xrow + kt * 4);
#pragma unroll
      for (int t = 0; t < 4; ++t) {
        // B layout (32-bit 4x16): lane n = col; v0/v1 = rows K = 4kt+{0,1}/{2,3}.
        const v2f b = *(const v2f*)(&sW1T[(t * 16 + col) * W1T_STRIDE + kt * 4 + hi * 2]);
        acc1[t] = __builtin_amdgcn_wmma_f32_16x16x4_f32(
            false, a, false, b, (short)0, acc1[t], false, false);
      }
    }

    // ReLU + C-layout -> [m][k] LDS staging (wave-private, no barrier needed;
    // DS ops are in-order within a wave, compiler inserts s_wait_dscnt).
#pragma unroll
    for (int t = 0; t < 4; ++t) {
#pragma unroll
      for (int r = 0; r < 8; ++r) {
        const float v = fmaxf(acc1[t][r], 0.0f);
        myH1[(r + 8 * hi) * H1_STRIDE + t * 16 + col] = v;   // h1[M][K]
      }
    }

    // ---------------- Layer 2: [16x64] @ [64x32] ----------------
    v8f acc2[2];
#pragma unroll
    for (int t = 0; t < 2; ++t) {
#pragma unroll
      for (int i = 0; i < 8; ++i) acc2[t][i] = b2v[t];
    }

#pragma unroll
    for (int kt = 0; kt < 16; ++kt) {
      const v2f a = *(const v2f*)(&myH1[col * H1_STRIDE + kt * 4 + hi * 2]);
#pragma unroll
      for (int t = 0; t < 2; ++t) {
        const v2f b = *(const v2f*)(&sW2T[(t * 16 + col) * W2T_STRIDE + kt * 4 + hi * 2]);
        acc2[t] = __builtin_amdgcn_wmma_f32_16x16x4_f32(
            false, a, false, b, (short)0, acc2[t], false, false);
      }
    }

    // ---------------- ReLU + mask + store ----------------
    // C layout: VGPR r holds M = r + 8*hi. One 8-byte mask load per lane covers
    // exactly rows base+8*hi .. base+8*hi+7; bit-extract per r.
    unsigned long long m64;
    if (full) {
      m64 = *(const unsigned long long*)(mask + base + 8 * hi);
    } else {
      m64 = 0ull;
#pragma unroll
      for (int i = 0; i < 8; ++i) {
        const int row = base + 8 * hi + i;
        const unsigned char mv = (row < E) ? mask[row] : (unsigned char)0;
        m64 |= (unsigned long long)(mv & 1u) << (8 * i);
      }
    }

#pragma unroll
    for (int r = 0; r < 8; ++r) {
      const int  row  = base + 8 * hi + r;
      const bool keep = ((m64 >> (8 * r)) & 1ull) != 0ull;
      if (row < E) {
        const float v0 = fmaxf(acc2[0][r], 0.0f);
        const float v1 = fmaxf(acc2[1][r], 0.0f);
        out[(size_t)row * 32 + col]      = keep ? v0 : 0.0f;
        out[(size_t)row * 32 + 16 + col] = keep ? v1 : 0.0f;
      }
    }
  }
}

extern "C" void kernel_launch(void* const* d_in, const int* in_sizes, int n_in,
                              void* d_out, int out_size, void* d_ws, size_t ws_size,
                              hipStream_t stream) {
  // setup_inputs() order: edge_attr, W1, b1, W2, b2, mask_edge
  const float*         x    = (const float*)d_in[0];
  const float*         W1   = (const float*)d_in[1];
  const float*         b1   = (const float*)d_in[2];
  const float*         W2   = (const float*)d_in[3];
  const float*         b2   = (const float*)d_in[4];
  const unsigned char* mask = (const unsigned char*)d_in[5];  // jax bool = 1 byte
  float* out = (float*)d_out;

  const int E      = in_sizes[0] / 32;
  const int nTiles = (E + 15) / 16;

  // Grid-stride over 16-edge tiles: 2048 blocks x 8 wave32 = 16384 waves,
  // ~6 tiles/wave at E=1.6M -> weight-transpose cost amortized, HBM saturated.
  int blocks = (nTiles + WAVES_PER_BLOCK - 1) / WAVES_PER_BLOCK;
  if (blocks > 2048) blocks = 2048;
  if (blocks < 1) blocks = 1;

  edge_mlp_wmma_f32<<<blocks, 256, 0, stream>>>(x, W1, b1, W2, b2, mask, out, E);
}